// GNO_51075751084700
// MI455X (gfx1250) — compile-verified
//
#include <hip/hip_runtime.h>
#include <hip/hip_bf16.h>
#include <math.h>

#define NNODES 50000
#define NEDGES 800000
#define F_IN   128
#define F_MID  256
#define F_OUT  128
#define LN_EPS 1e-5f

typedef __attribute__((ext_vector_type(2))) float v2f;
typedef __attribute__((ext_vector_type(8))) float v8f;

__device__ __forceinline__ float gelu_f(float x) {
  // exact erf GELU (matches torch F.gelu default / jax approximate=False)
  return 0.5f * x * (1.0f + erff(x * 0.70710678118654752440f));
}

// ---------------- degree / edge-norm precompute ----------------
__global__ void k_deg_init(float* __restrict__ deg, int n) {
  int i = blockIdx.x * blockDim.x + threadIdx.x;
  if (i < n) deg[i] = 1.0f;  // self-loop
}
__global__ void k_deg_edges(const int* __restrict__ dst, float* __restrict__ deg, int e) {
  int i = blockIdx.x * blockDim.x + threadIdx.x;
  if (i < e) atomicAdd(&deg[dst[i]], 1.0f);
}
__global__ void k_dinv(float* __restrict__ deg, int n) {
  int i = blockIdx.x * blockDim.x + threadIdx.x;
  if (i < n) deg[i] = rsqrtf(deg[i]);  // in place: deg -> dinv
}
__global__ void k_norm(const int* __restrict__ src, const int* __restrict__ dst,
                       const float* __restrict__ dinv, float* __restrict__ nrm, int e) {
  int i = blockIdx.x * blockDim.x + threadIdx.x;
  if (i < e) nrm[i] = dinv[src[i]] * dinv[dst[i]];
}

// ---------------- LayerNorm (optionally fused with GELU on the input) ----------------
// One wave32 per row, 8 rows per 256-thread block. F is 128 or 256.
template <int F, bool GELU>
__global__ __launch_bounds__(256) void k_ln(const float* __restrict__ X,
                                            const float* __restrict__ gamma,
                                            const float* __restrict__ beta,
                                            float* __restrict__ Y) {
  constexpr int NV = F / 128;  // float4 chunks per lane (1 or 2)
  const int lane = threadIdx.x & 31;
  const int wave = threadIdx.x >> 5;
  const size_t row = (size_t)blockIdx.x * 8 + wave;
  const float* xr = X + row * F;

  float4 v[NV];
  float s = 0.0f, ss = 0.0f;
#pragma unroll
  for (int i = 0; i < NV; ++i) {
    v[i] = *(const float4*)(xr + (size_t)(lane + i * 32) * 4);
    if (GELU) {
      v[i].x = gelu_f(v[i].x);
      v[i].y = gelu_f(v[i].y);
      v[i].z = gelu_f(v[i].z);
      v[i].w = gelu_f(v[i].w);
    }
    s  += v[i].x + v[i].y + v[i].z + v[i].w;
    ss += v[i].x * v[i].x + v[i].y * v[i].y + v[i].z * v[i].z + v[i].w * v[i].w;
  }
#pragma unroll
  for (int o = 16; o >= 1; o >>= 1) {
    s  += __shfl_xor(s,  o, 32);
    ss += __shfl_xor(ss, o, 32);
  }
  const float mu  = s * (1.0f / F);
  const float var = ss * (1.0f / F) - mu * mu;
  const float rstd = rsqrtf(var + LN_EPS);

  float* yr = Y + row * F;
#pragma unroll
  for (int i = 0; i < NV; ++i) {
    const int f = (lane + i * 32) * 4;
    const float4 g = *(const float4*)(gamma + f);
    const float4 b = *(const float4*)(beta + f);
    float4 o;
    o.x = (v[i].x - mu) * rstd * g.x + b.x;
    o.y = (v[i].y - mu) * rstd * g.y + b.y;
    o.z = (v[i].z - mu) * rstd * g.z + b.z;
    o.w = (v[i].w - mu) * rstd * g.w + b.w;
    *(float4*)(yr + f) = o;
  }
}

// ---------------- fp32 WMMA GEMM: C[M,NO] = A[M,K] @ B[K,NO] ----------------
// One wave per 16x16 output tile; 4 waves/block cover 4 adjacent N-tiles.
// Fragment layouts per CDNA5 ISA 7.12.2:
//   A 16x4 f32:  VGPR0 = K0 (lanes 0-15) / K2 (lanes 16-31), VGPR1 = K1 / K3
//   B 4x16 f32:  VGPR0 = row K0 / row K2, VGPR1 = row K1 / row K3 (rows striped across lanes)
//   C/D 16x16:   VGPRj = M=j (lanes 0-15) / M=j+8 (lanes 16-31)
template <int K, int NO>
__global__ __launch_bounds__(128) void k_gemm_wmma(const float* __restrict__ A,
                                                   const float* __restrict__ B,
                                                   float* __restrict__ C) {
  const int lane = threadIdx.x & 31;
  const int wave = threadIdx.x >> 5;
  const int row0 = blockIdx.x << 4;
  const int col0 = ((blockIdx.y << 2) + wave) << 4;
  const int m  = lane & 15;           // A row within tile / B-C-D column within tile
  const int kh = (lane >> 4) << 1;    // 0 or 2: K sub-offset for this lane half

  const float* arow = A + (size_t)(row0 + m) * K;
  const float* bcol = B + (size_t)col0 + m;

  v8f acc = {};
#pragma unroll 4
  for (int k0 = 0; k0 < K; k0 += 4) {
    v2f a, b;
    a.x = arow[k0 + kh];
    a.y = arow[k0 + kh + 1];
    const float* bp = bcol + (size_t)(k0 + kh) * NO;
    b.x = bp[0];
    b.y = bp[NO];
    acc = __builtin_amdgcn_wmma_f32_16x16x4_f32(false, a, false, b, (short)0, acc,
                                                false, false);
  }

  float* crow = C + (size_t)row0 * NO + col0 + m;
  const int mb = (lane >> 4) << 3;  // 0 or 8
#pragma unroll
  for (int j = 0; j < 8; ++j)
    crow[(size_t)(mb + j) * NO] = acc[j];
}

// ---------------- scatter init: Out[i,:] = H[i,:]*dinv[i]^2 + bias ----------------
template <int F>
__global__ void k_seg_init(const float* __restrict__ H, const float* __restrict__ dinv,
                           const float* __restrict__ bias, float* __restrict__ Out,
                           int nrows) {
  const size_t tid = (size_t)blockIdx.x * blockDim.x + threadIdx.x;
  const size_t total = (size_t)nrows * (F / 4);
  if (tid >= total) return;
  const size_t row = tid / (F / 4);
  const int f = (int)(tid % (F / 4)) * 4;
  float w = dinv[row];
  w *= w;  // self-loop norm = dinv^2
  const float4 h = *(const float4*)(H + row * F + f);
  const float4 b = *(const float4*)(bias + f);
  float4 o;
  o.x = h.x * w + b.x;
  o.y = h.y * w + b.y;
  o.z = h.z * w + b.z;
  o.w = h.w * w + b.w;
  *(float4*)(Out + row * F + f) = o;
}

// ---------------- edge scatter: Out[dst,:] += H[src,:]*norm[e] ----------------
// F/4 threads per edge, float4 gather + 4 f32 atomics.
template <int F>
__global__ void k_scatter(const float* __restrict__ H, const int* __restrict__ src,
                          const int* __restrict__ dst, const float* __restrict__ nrm,
                          float* __restrict__ Out, int e) {
  constexpr int TPE = F / 4;
  const size_t tid = (size_t)blockIdx.x * blockDim.x + threadIdx.x;
  const size_t eidx = tid / TPE;
  if (eidx >= (size_t)e) return;
  const int f = (int)(tid % TPE) * 4;
  const int s = src[eidx];
  const int d = dst[eidx];
  const float w = nrm[eidx];
  const float4 h = *(const float4*)(H + (size_t)s * F + f);
  float* o = Out + (size_t)d * F + f;
  atomicAdd(o + 0, h.x * w);
  atomicAdd(o + 1, h.y * w);
  atomicAdd(o + 2, h.z * w);
  atomicAdd(o + 3, h.w * w);
}

extern "C" void kernel_launch(void* const* d_in, const int* in_sizes, int n_in,
                              void* d_out, int out_size, void* d_ws, size_t ws_size,
                              hipStream_t stream) {
  (void)in_sizes; (void)n_in; (void)out_size; (void)ws_size;

  const float* x   = (const float*)d_in[0];  // [N,128]
  const float* W1  = (const float*)d_in[1];  // [128,256]
  const float* b1  = (const float*)d_in[2];  // [256]
  const float* g1  = (const float*)d_in[3];  // [128]
  const float* be1 = (const float*)d_in[4];  // [128]
  const float* W4  = (const float*)d_in[5];  // [256,128]
  const float* b4  = (const float*)d_in[6];  // [128]
  const float* g4  = (const float*)d_in[7];  // [256]
  const float* be4 = (const float*)d_in[8];  // [256]
  const int*   ei  = (const int*)d_in[9];    // [2,E]
  const int* src = ei;
  const int* dst = ei + NEDGES;
  float* out = (float*)d_out;

  // workspace carve-up (floats)
  float* ws   = (float*)d_ws;
  float* dinv = ws;                           // N         (deg -> dinv)
  float* nrm  = dinv + NNODES;                // E
  float* bufC = nrm + NEDGES;                 // N*128  ln(x), later h2
  float* bufA = bufC + (size_t)NNODES * 128;  // N*256  h1, later ln2 output
  float* bufB = bufA + (size_t)NNODES * 256;  // N*256  scatter accumulator s1

  const int T = 256;
  // degrees + per-edge norm
  k_deg_init<<<(NNODES + T - 1) / T, T, 0, stream>>>(dinv, NNODES);
  k_deg_edges<<<(NEDGES + T - 1) / T, T, 0, stream>>>(dst, dinv, NEDGES);
  k_dinv<<<(NNODES + T - 1) / T, T, 0, stream>>>(dinv, NNODES);
  k_norm<<<(NEDGES + T - 1) / T, T, 0, stream>>>(src, dst, dinv, nrm, NEDGES);

  // layer 1: LN -> GEMM (WMMA) -> init(self-loop+bias) -> edge scatter
  k_ln<128, false><<<NNODES / 8, 256, 0, stream>>>(x, g1, be1, bufC);
  k_gemm_wmma<128, 256><<<dim3(NNODES / 16, 4), 128, 0, stream>>>(bufC, W1, bufA);
  {
    size_t tot = (size_t)NNODES * (256 / 4);
    k_seg_init<256><<<(unsigned)((tot + T - 1) / T), T, 0, stream>>>(bufA, dinv, b1, bufB, NNODES);
    size_t st = (size_t)NEDGES * (256 / 4);
    k_scatter<256><<<(unsigned)((st + T - 1) / T), T, 0, stream>>>(bufA, src, dst, nrm, bufB, NEDGES);
  }

  // layer 2: fused GELU+LN -> GEMM (WMMA) -> init -> edge scatter into d_out
  k_ln<256, true><<<NNODES / 8, 256, 0, stream>>>(bufB, g4, be4, bufA);
  k_gemm_wmma<256, 128><<<dim3(NNODES / 16, 2), 128, 0, stream>>>(bufA, W4, bufC);
  {
    size_t tot = (size_t)NNODES * (128 / 4);
    k_seg_init<128><<<(unsigned)((tot + T - 1) / T), T, 0, stream>>>(bufC, dinv, b4, out, NNODES);
    size_t st = (size_t)NEDGES * (128 / 4);
    k_scatter<128><<<(unsigned)((st + T - 1) / T), T, 0, stream>>>(bufC, src, dst, nrm, out, NEDGES);
  }
}